// TrainableDelay_73452530696743
// MI455X (gfx1250) — compile-verified
//
#include <hip/hip_runtime.h>

// ---- problem geometry (fixed by reference) ----
#define T_STEPS   256
#define N_BATCH   16
#define C_CH      8
#define D_OUT     1024
#define D_IN      2
#define OI        (D_OUT * D_IN)               // 2048
#define SYN_TOTAL (N_BATCH * C_CH * D_OUT * D_IN)  // 262144 synapse columns
#define TPB       128                          // synapses per block
#define ROWS_PER_WAVE 64                       // 4 waves x 64 t-rows = 256

typedef __attribute__((ext_vector_type(4))) unsigned int u32x4;
typedef __attribute__((ext_vector_type(8))) int          i32x8;
typedef __attribute__((ext_vector_type(4))) int          i32x4;

__device__ __forceinline__ unsigned rotl32(unsigned x, unsigned r) {
  return (x << r) | (x >> (32u - r));
}

// threefry2x32, 20 rounds — exact match of JAX's PRNG core
__device__ __forceinline__ void threefry2x32(unsigned k0, unsigned k1,
                                             unsigned& x0, unsigned& x1) {
  const unsigned ks[3] = {k0, k1, k0 ^ k1 ^ 0x1BD11BDAu};
  const unsigned rotA[4] = {13u, 15u, 26u, 6u};
  const unsigned rotB[4] = {17u, 29u, 16u, 24u};
  x0 += ks[0];
  x1 += ks[1];
#pragma unroll
  for (int g = 0; g < 5; ++g) {
#pragma unroll
    for (int j = 0; j < 4; ++j) {
      unsigned r = (g & 1) ? rotB[j] : rotA[j];
      x0 += x1;
      x1 = rotl32(x1, r);
      x1 ^= x0;
    }
    x0 += ks[(g + 1) % 3];
    x1 += ks[(g + 2) % 3] + (unsigned)(g + 1);
  }
}

__global__ __launch_bounds__(TPB)
void trainable_delay_kernel(const float* __restrict__ in,
                            const float* __restrict__ delay,
                            float* __restrict__ out) {
  // LDS tile: [t][s_local], 256 x 128 floats = 128 KB.
  // Column reads/writes (fixed tid, varying t): addr/4 = t*128+tid
  // -> bank = tid mod 64, conflict-free for any t pattern across a wave32.
  __shared__ float tile[T_STEPS * TPB];

  const int tid  = threadIdx.x;
  const int wave = tid >> 5;
  const int s0   = blockIdx.x * TPB;   // first synapse of this block
  const int gs   = s0 + tid;           // this thread's global synapse index

  // ---------------- TDM: per-wave 2D tile DMA (global -> LDS) ----------------
  // Tensor: row-major, dim0 = synapse axis (stride-1), dim1 = t axis
  // (row stride SYN_TOTAL elements).  Each wave DMAs 64 rows x 128 cols.
  {
    unsigned lds_base = (unsigned)(unsigned long long)(uintptr_t)tile;
    unsigned lds_addr = lds_base + (unsigned)(wave * ROWS_PER_WAVE * TPB * 4);
    unsigned long long ga =
        (unsigned long long)(uintptr_t)in +
        4ull * ((unsigned long long)(wave * ROWS_PER_WAVE) * (unsigned long long)SYN_TOTAL +
                (unsigned long long)s0);

    // ---- D# group 0 (ISA 8.3): count=1, lds_addr, global_addr, type=2 ----
    u32x4 g0;
    g0.x = 1u;                                            // count=1, is_restore=0, gather off
    g0.y = lds_addr;                                      // bits 63:32
    g0.z = (unsigned)(ga & 0xFFFFFFFFull);                // global_addr[31:0]
    g0.w = ((unsigned)(ga >> 32) & 0x01FFFFFFu)           // global_addr[56:32]
           | (2u << 30);                                  // type=2 ("image")

    // ---- D# group 1 (ISA 8.4) ----
    const unsigned td0   = (unsigned)SYN_TOTAL;           // tensor_dim0
    const unsigned td1   = (unsigned)T_STEPS;             // tensor_dim1
    const unsigned tile0 = (unsigned)TPB;                 // tile_dim0
    const unsigned tile1 = (unsigned)ROWS_PER_WAVE;       // tile_dim1 (tile_dim2=0)
    const unsigned long long str0 = (unsigned long long)SYN_TOTAL; // dim0 stride
    i32x8 g1;
    g1[0] = (int)(2u << 16);                              // data_size=2 (4 bytes), no mcast/pad
    g1[1] = (int)((td0 & 0xFFFFu) << 16);                 // atomic_bar=0 | tensor_dim0[15:0]
    g1[2] = (int)((td0 >> 16) | ((td1 & 0xFFFFu) << 16)); // dim0[31:16] | dim1[15:0]
    g1[3] = (int)((td1 >> 16) | (tile0 << 16));           // dim1[31:16] | tile_dim0
    g1[4] = (int)(tile1);                                 // tile_dim1 | tile_dim2=0
    g1[5] = (int)(unsigned)(str0 & 0xFFFFFFFFull);        // dim0_stride[31:0]
    g1[6] = (int)(unsigned)((str0 >> 32) & 0xFFFFu);      // dim0_stride[47:32] | dim1_stride lo
    g1[7] = 0;                                            // dim1_stride hi (unused for 2D)

    i32x4 gz4 = {0, 0, 0, 0};                             // groups 2/3 unused (2D tensor)
    i32x8 gz8 = {0, 0, 0, 0, 0, 0, 0, 0};                 // extra operand (clang-23 form): zero

    __builtin_amdgcn_tensor_load_to_lds(g0, g1, gz4, gz4, gz8, 0);
    __builtin_amdgcn_s_wait_tensorcnt(0);
  }
  __syncthreads();   // make all 4 waves' DMA'd tiles visible

  // ------- pass 1: sigmoid in place + argmax of sigmoid over t -------
  // Each thread owns column `tid` exclusively from here on -> no more barriers.
  // jnp.argmax keeps first occurrence -> strict '>' comparison.
  // Reference argmax is over sigmoid(x), so compare the sigmoid values.
  float best = -1.0f;   // sigmoid > 0 always
  int   amax = 0;
#pragma unroll 8
  for (int t = 0; t < T_STEPS; ++t) {
    float x  = tile[t * TPB + tid];
    float sg = __builtin_amdgcn_rcpf(1.0f + __expf(-x));
    tile[t * TPB + tid] = sg;
    if (sg > best) { best = sg; amax = t; }
  }

  // ---------------- delay: threefry bernoulli round + clamp ----------------
  float dlat   = delay[gs & (OI - 1)];     // broadcast over (N,C): index = gs % 2048
  float dfloor = floorf(dlat);
  float p      = dlat - dfloor;

  // jax.random.bernoulli(key(42), p, shape=(16,8,1024,2)):
  //   counts = iota(262144); x0 = counts[:131072], x1 = counts[131072:]
  //   out = concat(threefry(key, x0, x1))
  const unsigned half = (unsigned)(SYN_TOTAL / 2);
  const unsigned us   = (unsigned)gs;
  unsigned x0 = (us < half) ? us : (us - half);
  unsigned x1 = x0 + half;
  threefry2x32(0u, 42u, x0, x1);
  unsigned bits = (us < half) ? x0 : x1;
  // JAX uniform in [0,1): bitcast(bits>>9 | 1.0f) - 1.0f
  union { unsigned u; float f; } cvt;
  cvt.u = (bits >> 9) | 0x3F800000u;
  float u01 = cvt.f - 1.0f;

  int dint = (int)dfloor + ((u01 < p) ? 1 : 0);
  int dmax = (T_STEPS - 1) - amax;
  dint = (dint < dmax) ? dint : dmax;

  // ---------------- pass 2: circular shift + streaming store ----------------
  // Output is write-once dead data: non-temporal stores keep it out of L2
  // so concurrent blocks' TDM input streams keep the cache.
  float* __restrict__ outp = out + gs;
#pragma unroll 8
  for (int t = 0; t < T_STEPS; ++t) {
    int src = (t - dint) & (T_STEPS - 1);
    float sg = tile[src * TPB + tid];
    __builtin_nontemporal_store(sg, &outp[(size_t)t * SYN_TOTAL]);
  }
}

extern "C" void kernel_launch(void* const* d_in, const int* in_sizes, int n_in,
                              void* d_out, int out_size, void* d_ws, size_t ws_size,
                              hipStream_t stream) {
  (void)in_sizes; (void)n_in; (void)out_size; (void)d_ws; (void)ws_size;
  const float* in    = (const float*)d_in[0];
  const float* delay = (const float*)d_in[1];
  float*       outp  = (float*)d_out;

  dim3 grid(SYN_TOTAL / TPB);   // 2048 blocks
  dim3 block(TPB);              // 128 threads = 4 waves
  trainable_delay_kernel<<<grid, block, 0, stream>>>(in, delay, outp);
}